// Multi_Head_Attention_45569603010923
// MI455X (gfx1250) — compile-verified
//
#include <hip/hip_runtime.h>

typedef __attribute__((ext_vector_type(2))) float v2f;
typedef __attribute__((ext_vector_type(8))) float v8f;

#define BATCH 8
#define HNUM  12
#define BH    96      // BATCH*HNUM
#define NSEQ  1024
#define DH    64
#define ENUM  768
#define KVPITCH 68    // LDS row pitch (floats): stride%64==4 -> conflict-free A-frag reads,
                      // 272B row stride keeps 16B alignment for async B128 fills

__device__ __forceinline__ v8f wmma_f32_k4(v2f a, v2f b, v8f c) {
  // D = A(16x4,f32) * B(4x16,f32) + C(16x16,f32)
  return __builtin_amdgcn_wmma_f32_16x16x4_f32(
      /*neg_a=*/false, a, /*neg_b=*/false, b,
      /*c_mod=*/(short)0, c, /*reuse_a=*/false, /*reuse_b=*/false);
}

// generic pointer to LDS: low 32 bits are the wave-relative LDS byte address
__device__ __forceinline__ uint32_t lds_addr32(const void* p) {
  return (uint32_t)(uintptr_t)p;
}

// ---------------------------------------------------------------------------
// Kernel 1: per-head QKV projection.  xh[n,d] = x[b, n, d*HNUM + h]
// Q/K/V[n,e] = sum_d xh[n,d] * W[e,d] + bias[e]   (16x64 tile per wave)
// ---------------------------------------------------------------------------
__device__ __forceinline__ void proj_one(const v2f (&a)[16],
                                         const float* __restrict__ W,
                                         const float* __restrict__ bias,
                                         float* __restrict__ o,  // [NSEQ][DH]
                                         int lane15, int hi, int n0) {
  #pragma unroll
  for (int c = 0; c < 4; ++c) {
    const float bcol = bias[c * 16 + lane15];   // bias depends on N only
    v8f acc;
    #pragma unroll
    for (int r = 0; r < 8; ++r) acc[r] = bcol;
    const float* Wrow = W + (c * 16 + lane15) * DH + 2 * hi;
    #pragma unroll
    for (int k = 0; k < 16; ++k) {
      v2f bf;
      bf.x = Wrow[4 * k];       // B[K=2hi  , N=lane15] = W[e, 4k+2hi]
      bf.y = Wrow[4 * k + 1];   // B[K=2hi+1, N=lane15]
      acc = wmma_f32_k4(a[k], bf, acc);
    }
    #pragma unroll
    for (int r = 0; r < 8; ++r) {
      const int n = n0 + r + 8 * hi;            // C: M = r + 8*(lane/16)
      o[(size_t)n * DH + c * 16 + lane15] = acc[r];
    }
  }
}

__global__ __launch_bounds__(256)
void qkv_proj_kernel(const float* __restrict__ x,
                     const float* __restrict__ Wq, const float* __restrict__ bq,
                     const float* __restrict__ Wk, const float* __restrict__ bk,
                     const float* __restrict__ Wv, const float* __restrict__ bv,
                     float* __restrict__ Qws, float* __restrict__ Kws,
                     float* __restrict__ Vws) {
  const int wave   = threadIdx.x >> 5;
  const int lane   = threadIdx.x & 31;
  const int lane15 = lane & 15;
  const int hi     = lane >> 4;
  const int bh     = blockIdx.y;
  const int b      = bh / HNUM;
  const int h      = bh % HNUM;
  const int n0     = blockIdx.x * 128 + wave * 16;

  // A fragments: A[M=lane15, K=d], d = 4k + v + 2*hi (strided head slice)
  v2f a[16];
  const float* xrow = x + ((size_t)b * NSEQ + n0 + lane15) * ENUM + h;
  #pragma unroll
  for (int k = 0; k < 16; ++k) {
    const int d0 = 4 * k + 2 * hi;
    a[k].x = xrow[(size_t)d0 * HNUM];
    a[k].y = xrow[(size_t)(d0 + 1) * HNUM];
  }

  const size_t head_off = (size_t)bh * NSEQ * DH;
  proj_one(a, Wq, bq, Qws + head_off, lane15, hi, n0);
  proj_one(a, Wk, bk, Kws + head_off, lane15, hi, n0);
  proj_one(a, Wv, bv, Vws + head_off, lane15, hi, n0);
}

// ---------------------------------------------------------------------------
// Kernel 2: flash attention, transposed formulation.
//   S^T = K_tile @ Q^T   (A = K tile from LDS, B = Q frags, C = S^T)
//   O^T += V^T @ P^T     (A = V^T from LDS,   B = P^T via LDS slab)
// K/V tiles staged into LDS once per block with async-to-LDS, double-buffered.
// ---------------------------------------------------------------------------
__global__ __launch_bounds__(256)
void flash_attn_kernel(const float* __restrict__ Qws,
                       const float* __restrict__ Kws,
                       const float* __restrict__ Vws,
                       float* __restrict__ out) {
  __shared__ float kbuf[2][16 * KVPITCH];
  __shared__ float vbuf[2][16 * KVPITCH];
  __shared__ float pslab[8][16][17];    // per-wave P^T round-trip tile

  const int tid    = threadIdx.x;
  const int wave   = tid >> 5;
  const int lane   = tid & 31;
  const int lane15 = lane & 15;
  const int hi     = lane >> 4;
  const int bh     = blockIdx.y;
  const int n0     = blockIdx.x * 128 + wave * 16;

  const float scale = 0.03608439182435161f;  // 1/sqrt(768)

  const float* Qh = Qws + (size_t)bh * NSEQ * DH;
  const float* Kh = Kws + (size_t)bh * NSEQ * DH;
  const float* Vh = Vws + (size_t)bh * NSEQ * DH;

  // cooperative tile fill: thread -> (row, 16B segment); one B128 per matrix
  const int frow = tid >> 4;          // 0..15
  const int fseg = (tid & 15) << 2;   // float offset 0,4,...,60

  // Q B-fragments (scale folded in): B[K=v+2hi, N=lane15] = Q[n, 4k+v+2hi]
  v2f qb[16];
  const float* qrow = Qh + (size_t)(n0 + lane15) * DH + 2 * hi;
  #pragma unroll
  for (int k = 0; k < 16; ++k) {
    qb[k].x = qrow[4 * k] * scale;
    qb[k].y = qrow[4 * k + 1] * scale;
  }

  v8f o[4];
  #pragma unroll
  for (int c = 0; c < 4; ++c)
    #pragma unroll
    for (int r = 0; r < 8; ++r) o[c][r] = 0.0f;
  float mR = -1e30f, lR = 0.0f;       // per query-row stats (row = lane15)

  // prologue: fill buffer 0 for tile j=0
  {
    const uint32_t ko = lds_addr32(&kbuf[0][frow * KVPITCH + fseg]);
    const uint32_t vo = lds_addr32(&vbuf[0][frow * KVPITCH + fseg]);
    const float* gk = Kh + (size_t)frow * DH + fseg;
    const float* gv = Vh + (size_t)frow * DH + fseg;
    asm volatile("global_load_async_to_lds_b128 %0, %2, off\n\t"
                 "global_load_async_to_lds_b128 %1, %3, off"
                 :: "v"(ko), "v"(vo), "v"(gk), "v"(gv) : "memory");
  }
  asm volatile("s_wait_asynccnt 0" ::: "memory");
  __syncthreads();

  for (int j = 0; j < NSEQ / 16; ++j) {
    const int cur = j & 1;

    // prefetch tile j+1 into the other buffer (overlaps with compute)
    if (j + 1 < NSEQ / 16) {
      const int nxt = cur ^ 1;
      const int m1 = (j + 1) * 16;
      const uint32_t ko = lds_addr32(&kbuf[nxt][frow * KVPITCH + fseg]);
      const uint32_t vo = lds_addr32(&vbuf[nxt][frow * KVPITCH + fseg]);
      const float* gk = Kh + (size_t)(m1 + frow) * DH + fseg;
      const float* gv = Vh + (size_t)(m1 + frow) * DH + fseg;
      asm volatile("global_load_async_to_lds_b128 %0, %2, off\n\t"
                   "global_load_async_to_lds_b128 %1, %3, off"
                   :: "v"(ko), "v"(vo), "v"(gk), "v"(gv) : "memory");
    }

    // S^T = K_tile @ Q^T : A[M=lane15(m), K=v+2hi] = K[m, 4k+v+2hi]
    v8f S;
    #pragma unroll
    for (int r = 0; r < 8; ++r) S[r] = 0.0f;
    const float* kb = &kbuf[cur][0];
    #pragma unroll
    for (int k = 0; k < 16; ++k) {
      v2f a;
      a.x = kb[lane15 * KVPITCH + 4 * k + 2 * hi];
      a.y = kb[lane15 * KVPITCH + 4 * k + 2 * hi + 1];
      S = wmma_f32_k4(a, qb[k], S);
    }

    // online softmax along m: 16 scores of query row lane15 live in this lane
    // pair (hi=0 -> m=r, hi=1 -> m=r+8); combine halves with one shfl_xor(16).
    float loc = S[0];
    #pragma unroll
    for (int r = 1; r < 8; ++r) loc = fmaxf(loc, S[r]);
    loc = fmaxf(loc, __shfl_xor(loc, 16, 32));
    const float mnew  = fmaxf(mR, loc);
    const float alpha = __expf(mR - mnew);
    float p[8];
    float sum = 0.0f;
    #pragma unroll
    for (int r = 0; r < 8; ++r) { p[r] = __expf(S[r] - mnew); sum += p[r]; }
    sum += __shfl_xor(sum, 16, 32);
    lR = lR * alpha + sum;
    mR = mnew;
    #pragma unroll
    for (int c = 0; c < 4; ++c)
      #pragma unroll
      for (int r = 0; r < 8; ++r) o[c][r] *= alpha;

    // P^T: C layout -> per-wave LDS slab -> B layout (in-order DS within wave)
    #pragma unroll
    for (int r = 0; r < 8; ++r) pslab[wave][r + 8 * hi][lane15] = p[r];
    v2f pb[4];
    #pragma unroll
    for (int kk = 0; kk < 4; ++kk) {
      pb[kk].x = pslab[wave][4 * kk + 2 * hi][lane15];
      pb[kk].y = pslab[wave][4 * kk + 2 * hi + 1][lane15];
    }

    // O^T += V^T @ P^T : A[M=lane15(d'), K=v+2hi] = V[4kk+v+2hi, c*16+d']
    const float* vb = &vbuf[cur][0];
    #pragma unroll
    for (int c = 0; c < 4; ++c) {
      #pragma unroll
      for (int kk = 0; kk < 4; ++kk) {
        v2f a;
        a.x = vb[(4 * kk + 2 * hi) * KVPITCH + c * 16 + lane15];
        a.y = vb[(4 * kk + 2 * hi + 1) * KVPITCH + c * 16 + lane15];
        o[c] = wmma_f32_k4(a, pb[kk], o[c]);
      }
    }

    // publish next tile / protect current buffers
    asm volatile("s_wait_asynccnt 0" ::: "memory");
    __syncthreads();
  }

  // normalize and store: O^T element (M=d'=r+8hi, N=n=lane15)
  // per lane: 8 consecutive dwords per c-tile -> coalesced wide stores
  const float inv = 1.0f / lR;
  float* orow = out + ((size_t)bh * NSEQ + n0 + lane15) * DH;
  #pragma unroll
  for (int c = 0; c < 4; ++c)
    #pragma unroll
    for (int r = 0; r < 8; ++r)
      orow[c * 16 + r + 8 * hi] = o[c][r] * inv;
}

// ---------------------------------------------------------------------------
extern "C" void kernel_launch(void* const* d_in, const int* in_sizes, int n_in,
                              void* d_out, int out_size, void* d_ws, size_t ws_size,
                              hipStream_t stream) {
  const float* x  = (const float*)d_in[0];
  const float* Wq = (const float*)d_in[1];
  const float* bq = (const float*)d_in[2];
  const float* Wk = (const float*)d_in[3];
  const float* bk = (const float*)d_in[4];
  const float* Wv = (const float*)d_in[5];
  const float* bv = (const float*)d_in[6];
  float* out = (float*)d_out;

  const size_t per = (size_t)BH * NSEQ * DH;   // 6,291,456 floats (~25.2 MB)
  float* Qws = (float*)d_ws;
  float* Kws = Qws + per;
  float* Vws = Kws + per;

  dim3 grid(NSEQ / 128, BH);   // (8, 96)
  dim3 block(256);             // 8 waves

  qkv_proj_kernel<<<grid, block, 0, stream>>>(x, Wq, bq, Wk, bk, Wv, bv,
                                              Qws, Kws, Vws);
  flash_attn_kernel<<<grid, block, 0, stream>>>(Qws, Kws, Vws, out);
}